// ALS_net_76699525972150
// MI455X (gfx1250) — compile-verified
//
#include <hip/hip_runtime.h>

typedef __attribute__((ext_vector_type(2))) float v2f;
typedef __attribute__((ext_vector_type(8))) float v8f;

#define KDIM 128
#define TILE 16                                  // batch elements per WMMA tile
#define SPLITK 2                                 // waves cooperating per tile
#define WAVES_PER_BLOCK 8
#define TILES_PER_BLOCK (WAVES_PER_BLOCK / SPLITK)   // 4 tiles per 256-thread block

// Two wave32s per 16-batch tile: each accumulates diag( A(16x64) * B(64x16) )
// over half of K with V_WMMA_F32_16X16X4_F32, then pairwise-reduce via LDS.
__global__ __launch_bounds__(WAVES_PER_BLOCK * 32)
void ALS_gather_dot_wmma(const float* __restrict__ U,       // [USERS, 128] row-major
                         const float* __restrict__ G,       // [128, GOODS] row-major
                         const long long* __restrict__ loc, // [BATCH, 2] int64
                         float* __restrict__ out,           // [BATCH] f32
                         int goods_num)
{
    __shared__ float part[TILES_PER_BLOCK][TILE];

    const int lane = threadIdx.x & 31;
    const int wave = threadIdx.x >> 5;
    const int t    = wave >> 1;                  // tile slot within block
    const int p    = wave & 1;                   // which K half this wave owns
    const int base = (blockIdx.x * TILES_PER_BLOCK + t) * TILE;

    const int m  = lane & 15;                    // A row / B col served by lane
    const int kb = (lane >> 4) << 1;             // 0 (lanes 0-15) or 2 (lanes 16-31)

    const long long* lp = loc + (size_t)(base + m) * 2;
    const unsigned urow = (unsigned)lp[0];
    const unsigned gcol = (unsigned)lp[1];

    const float*   uptr  = U + (size_t)urow * KDIM;
    const unsigned gstep = (unsigned)goods_num;

    const int kbeg = p * (KDIM / SPLITK);
    const int kend = kbeg + (KDIM / SPLITK);

    v8f c = {};
#pragma unroll 8
    for (int k0 = kbeg; k0 < kend; k0 += 4) {
        // A 16x4: lanes 0-15 carry K={k0,k0+1}, lanes 16-31 K={k0+2,k0+3}
        v2f a = *(const v2f*)(uptr + k0 + kb);
        // B 4x16: same half-lane K split; 32-bit element offsets -> SADDR form
        unsigned goff = (unsigned)(k0 + kb) * gstep + gcol;
        v2f b;
        b.x = G[goff];
        b.y = G[goff + gstep];
        c = __builtin_amdgcn_wmma_f32_16x16x4_f32(false, a, false, b,
                                                  (short)0, c, false, false);
    }

    // Branchless diagonal extraction:
    // (m,m) lives in VGPR m @ lane m (m<8) and VGPR m-8 @ lane m+16 (m>=8)
    const bool lo  = (lane < 8);
    const bool hi  = (lane >= 24);
    const int  sel = lo ? lane : (lane - 24);    // c[] index (valid only if lo||hi)
    float val = c[0];
#pragma unroll
    for (int r = 1; r < 8; ++r) val = (sel == r) ? c[r] : val;
    const int mm = lo ? lane : (lane - 16);      // diagonal index within tile

    if (p == 0 && (lo || hi)) part[t][mm] = val;         // 16 partials -> LDS
    __syncthreads();
    if (p == 1 && (lo || hi)) out[base + mm] = val + part[t][mm];
}

extern "C" void kernel_launch(void* const* d_in, const int* in_sizes, int n_in,
                              void* d_out, int out_size, void* d_ws, size_t ws_size,
                              hipStream_t stream) {
    const float*     U   = (const float*)d_in[0];      // user_matrix  [USERS,128]
    const float*     G   = (const float*)d_in[1];      // goods_matrix [128,GOODS]
    const long long* loc = (const long long*)d_in[2];  // location     [BATCH,2] int64
    float*           out = (float*)d_out;

    const int goods_num = in_sizes[1] / KDIM;
    const int batch     = in_sizes[2] / 2;

    const int tiles  = batch / TILE;                   // 1024
    const int blocks = tiles / TILES_PER_BLOCK;        // 256 blocks x 8 waves
    ALS_gather_dot_wmma<<<blocks, WAVES_PER_BLOCK * 32, 0, stream>>>(
        U, G, loc, out, goods_num);
}